// VectorQuantizerEMA_12000138625223
// MI455X (gfx1250) — compile-verified
//
#include <hip/hip_runtime.h>
#include <hip/hip_bf16.h>
#include <math.h>

typedef __attribute__((ext_vector_type(2))) float v2f;
typedef __attribute__((ext_vector_type(8))) float v8f;

#define NUM_EMB 512
#define EMB_DIM 64
#define DHW     16384      // 16*32*32
#define NROWS   131072     // 8 * DHW
#define DECAYF  0.99f
#define OMDF    0.01f
#define EPSF    1e-6f

// workspace layout (float offsets)
#define WS_COUNTS 0
#define WS_EMBSUM 512
#define WS_LOSS   (512 + 512*64)   // 33280
#define WS_E2     (WS_LOSS + 1)    // 33281

// output layout (float offsets, tuple order)
#define OUT_Q      0
#define OUT_LOSS   8388608
#define OUT_IDX    8388609
#define OUT_PERP   8519681
#define OUT_USED   8519682
#define OUT_NEWEMB 8519683
#define OUT_NEWCS  8552451
#define OUT_NEWAVG 8552963

#define WAVES_PER_BLOCK 8

__global__ void vq_zero(float* __restrict__ ws) {
    int i = blockIdx.x * blockDim.x + threadIdx.x;
    if (i < 512 + 512 * 64 + 1) ws[i] = 0.0f;   // counts + embed_sum + loss
}

__global__ void vq_e2(const float* __restrict__ emb, float* __restrict__ ws) {
    int m = blockIdx.x * blockDim.x + threadIdx.x;
    if (m < NUM_EMB) {
        float s = 0.0f;
        #pragma unroll 8
        for (int c = 0; c < EMB_DIM; ++c) { float e = emb[m * EMB_DIM + c]; s += e * e; }
        ws[WS_E2 + m] = s;
    }
}

__launch_bounds__(256)
__global__ void vq_main(const float* __restrict__ z,
                        const float* __restrict__ emb,
                        float* __restrict__ ws,
                        float* __restrict__ out) {
    __shared__ float flatLDS[WAVES_PER_BLOCK][16][EMB_DIM]; // 32 KB
    __shared__ float x2LDS[WAVES_PER_BLOCK][16];
    __shared__ int   idxLDS[WAVES_PER_BLOCK][16];
    __shared__ float keyLDS[WAVES_PER_BLOCK][16];

    const int lane = threadIdx.x & 31;
    const int wave = threadIdx.x >> 5;
    const int tile = blockIdx.x * WAVES_PER_BLOCK + wave;  // 0..8191, exact
    const int row0 = tile << 4;
    const int b    = row0 / DHW;
    const int r    = row0 - b * DHW;
    const int hl   = lane >> 4;     // half-wave: 0 -> lanes 0-15, 1 -> 16-31
    const int l16  = lane & 15;

    // ---- stage 16x64 flat tile: z is [B,C,D,H,W]; rows are consecutive in DHW ----
    const float* zb = z + (size_t)b * EMB_DIM * DHW + r;
    #pragma unroll 4
    for (int t = 0; t < 32; ++t) {
        int c = 2 * t + hl;
        flatLDS[wave][l16][c] = zb[(size_t)c * DHW + l16];  // 16-float coalesced per half
    }
    __syncthreads();

    // ---- per-row ||x||^2 ----
    if (hl == 0) {
        float s = 0.0f;
        #pragma unroll 8
        for (int c = 0; c < EMB_DIM; ++c) { float v = flatLDS[wave][l16][c]; s += v * v; }
        x2LDS[wave][l16] = s;
    }

    // ---- A fragments, ISA 32-bit 16x4 layout: lanes 0-15 hold K={4i,4i+1}, 16-31 K={4i+2,4i+3} ----
    v2f afrag[16];
    #pragma unroll
    for (int i = 0; i < 16; ++i) {
        int cb = 4 * i + 2 * hl;
        afrag[i].x = flatLDS[wave][l16][cb];
        afrag[i].y = flatLDS[wave][l16][cb + 1];
    }

    const float* e2 = ws + WS_E2;
    float minKey[8];
    int   minIdx[8];
    #pragma unroll
    for (int v = 0; v < 8; ++v) { minKey[v] = 3.4e38f; minIdx[v] = 0; }

    // issue a 16-load clause for tile j's B fragments (+ its ||e||^2) into dst
    auto loadB = [&](int j, v2f* dst) -> float {
        const int n = (j << 4) + l16;
        const float* eb = emb + n * EMB_DIM + 2 * hl;
        #pragma unroll
        for (int i = 0; i < 16; ++i) dst[i] = *(const v2f*)(eb + 4 * i);
        return e2[n];
    };
    // 16-deep chained fp32 WMMA, then fold into the running argmin keys
    auto compute = [&](int j, const v2f* bf, float e2n) {
        v8f acc = {};
        #pragma unroll
        for (int i = 0; i < 16; ++i)
            acc = __builtin_amdgcn_wmma_f32_16x16x4_f32(
                false, afrag[i], false, bf[i], (short)0, acc, false, false);
        const int n = (j << 4) + l16;
        #pragma unroll
        for (int v = 0; v < 8; ++v) {
            float key = fmaf(-2.0f, acc[v], e2n);   // d2 minus row-constant ||x||^2
            if (key < minKey[v]) { minKey[v] = key; minIdx[v] = n; }
        }
    };

    // ---- sweep 512 codes, double-buffered: loads for j+1 overlap WMMA chain for j ----
    v2f bufA[16], bufB[16];
    float eA = loadB(0, bufA);
    for (int j = 0; j < NUM_EMB / 16; j += 2) {
        float eB = loadB(j + 1, bufB);
        compute(j, bufA, eA);
        eA = (j + 2 < NUM_EMB / 16) ? loadB(j + 2, bufA) : 0.0f;
        compute(j + 1, bufB, eB);
    }

    // ---- argmin across the 16 lanes of each half (C layout: half h holds M = v + 8h) ----
    #pragma unroll
    for (int v = 0; v < 8; ++v) {
        float k = minKey[v]; int ix = minIdx[v];
        #pragma unroll
        for (int m = 1; m < 16; m <<= 1) {
            float ok = __shfl_xor(k, m, 32);
            int   oi = __shfl_xor(ix, m, 32);
            if (ok < k || (ok == k && oi < ix)) { k = ok; ix = oi; }
        }
        if (l16 == 0) {
            int row = v + 8 * hl;
            idxLDS[wave][row] = ix;
            keyLDS[wave][row] = k;
        }
    }
    __syncthreads();

    // ---- per-row outputs: indices, counts, loss ----
    if (hl == 0) {
        const int code = idxLDS[wave][l16];
        out[OUT_IDX + row0 + l16] = (float)code;
        atomicAdd(ws + WS_COUNTS + code, 1.0f);
        float d2 = fmaxf(x2LDS[wave][l16] + keyLDS[wave][l16], 0.0f);
        #pragma unroll
        for (int m = 1; m < 16; m <<= 1) d2 += __shfl_xor(d2, m, 32);
        if (l16 == 0) atomicAdd(ws + WS_LOSS, d2);
    }

    // ---- quantized_st (== gathered codebook rows) + embed_sum segment-sum ----
    float* embsum = ws + WS_EMBSUM;
    float* qb = out + OUT_Q + (size_t)b * EMB_DIM * DHW + r;
    #pragma unroll 2
    for (int t = 0; t < 32; ++t) {
        int c    = 2 * t + hl;
        int code = idxLDS[wave][l16];
        qb[(size_t)c * DHW + l16] = emb[code * EMB_DIM + c];
        atomicAdd(embsum + code * EMB_DIM + c, flatLDS[wave][l16][c]);
    }
}

__launch_bounds__(512)
__global__ void vq_finalize(const float* __restrict__ cluster_size,
                            const float* __restrict__ embedding_avg,
                            const float* __restrict__ ws,
                            float* __restrict__ out) {
    __shared__ float red[512];
    const int m = threadIdx.x;

    const float counts = ws[WS_COUNTS + m];
    const float new_cs = cluster_size[m] * DECAYF + OMDF * counts;
    out[OUT_NEWCS + m] = new_cs;

    // n = sum(new_cluster_size)
    red[m] = new_cs; __syncthreads();
    for (int s = 256; s > 0; s >>= 1) { if (m < s) red[m] += red[m + s]; __syncthreads(); }
    const float n = red[0]; __syncthreads();

    // perplexity accumulator
    const float p = counts / (float)NROWS;
    red[m] = p * logf(p + 1e-10f); __syncthreads();
    for (int s = 256; s > 0; s >>= 1) { if (m < s) red[m] += red[m + s]; __syncthreads(); }
    const float plog = red[0]; __syncthreads();

    // used codes from OLD cluster_size
    red[m] = (cluster_size[m] > 1e-5f) ? 1.0f : 0.0f; __syncthreads();
    for (int s = 256; s > 0; s >>= 1) { if (m < s) red[m] += red[m + s]; __syncthreads(); }
    const float used = red[0];

    const float cs    = (new_cs + EPSF) / (n + NUM_EMB * EPSF) * n;
    const float denom = cs + EPSF;
    #pragma unroll 4
    for (int c = 0; c < EMB_DIM; ++c) {
        float ea = embedding_avg[m * EMB_DIM + c] * DECAYF
                 + OMDF * ws[WS_EMBSUM + m * EMB_DIM + c];
        out[OUT_NEWAVG + m * EMB_DIM + c] = ea;
        out[OUT_NEWEMB + m * EMB_DIM + c] = ea / denom;
    }

    if (m == 0) {
        out[OUT_LOSS] = ws[WS_LOSS] / 8388608.0f;  // mean over B*C*D*H*W
        out[OUT_PERP] = expf(-plog);
        out[OUT_USED] = used / (float)NUM_EMB;
    }
}

extern "C" void kernel_launch(void* const* d_in, const int* in_sizes, int n_in,
                              void* d_out, int out_size, void* d_ws, size_t ws_size,
                              hipStream_t stream) {
    const float* z   = (const float*)d_in[0];
    const float* emb = (const float*)d_in[1];
    const float* cs  = (const float*)d_in[2];
    const float* avg = (const float*)d_in[3];
    float* out = (float*)d_out;
    float* ws  = (float*)d_ws;

    vq_zero<<<(512 + 512 * 64 + 1 + 255) / 256, 256, 0, stream>>>(ws);
    vq_e2<<<2, 256, 0, stream>>>(emb, ws);
    vq_main<<<NROWS / 16 / WAVES_PER_BLOCK, 256, 0, stream>>>(z, emb, ws, out);
    vq_finalize<<<1, 512, 0, stream>>>(cs, avg, ws, out);
}